// TestGraphNetwork_82231443849935
// MI455X (gfx1250) — compile-verified
//
#include <hip/hip_runtime.h>

typedef __attribute__((ext_vector_type(2))) float v2f;
typedef __attribute__((ext_vector_type(8))) float v8f;

#define C_DIM 128
#define H_DIM 256

// Zero the x1 accumulator and emit y / batch_0 as floats into d_out[0..2N).
__global__ void init_zero_copy(float* __restrict__ x1, long ec,
                               const int* __restrict__ y,
                               const int* __restrict__ b0,
                               float* __restrict__ out, int n) {
    long i = (long)blockIdx.x * blockDim.x + threadIdx.x;
    if (i < ec) x1[i] = 0.0f;
    if (i < n) {
        out[i]     = (float)y[i];
        out[n + i] = (float)b0[i];
    }
}

// One wave32 per nonzero: lane handles 4 channels (32*4 = 128), scatter-add
// into x1[row]. x (51MB) is L2-resident so the random gathers stay on-chip.
__global__ void scatter_hyperedges(const float* __restrict__ x,
                                   const int* __restrict__ rows,
                                   const int* __restrict__ cols,
                                   const float* __restrict__ vals,
                                   float* __restrict__ x1, int nnz) {
    int wave = blockIdx.x * (blockDim.x >> 5) + (threadIdx.x >> 5);
    int lane = threadIdx.x & 31;
    if (wave >= nnz) return;
    int   r = rows[wave];
    int   c = cols[wave];
    float v = vals[wave];
    const float4* src = (const float4*)(x + (long)c * C_DIM);
    float4 g = src[lane];
    float* dst = x1 + (long)r * C_DIM + lane * 4;
    atomicAdd(dst + 0, g.x * v);
    atomicAdd(dst + 1, g.y * v);
    atomicAdd(dst + 2, g.z * v);
    atomicAdd(dst + 3, g.w * v);
}

// out = relu(A[M,128] @ W[128,256] + bias), one wave32 per 16x16 tile,
// K swept 4 at a time with V_WMMA_F32_16X16X4_F32 (full fp32 precision).
__global__ void gemm_bias_relu(const float* __restrict__ A,
                               const float* __restrict__ W,
                               const float* __restrict__ bias,
                               float* __restrict__ out, int M) {
    const int  lane = threadIdx.x & 31;
    const int  wave = threadIdx.x >> 5;
    const long tile = (long)blockIdx.x * (blockDim.x >> 5) + wave;
    const int  ntiles = H_DIM / 16;
    const long mt = tile / ntiles;
    const int  nt = (int)(tile % ntiles);
    if (mt * 16 >= M) return;   // wave-uniform; M is a multiple of 16 so EXEC stays all-ones

    const int rc = lane & 15;              // A row within tile / B-C-D column within tile
    const int kh = (lane >> 4) << 1;       // lanes 0-15 hold K pair {0,1}; lanes 16-31 hold {2,3}
    const float* arow = A + (mt * 16 + rc) * C_DIM;
    const float* wcol = W + nt * 16 + rc;
    __builtin_prefetch(arow, 0, 1);
    __builtin_prefetch(wcol, 0, 1);

    v8f acc = {};
    #pragma unroll 4
    for (int k = 0; k < C_DIM; k += 4) {
        v2f a, b;
        a.x = arow[k + kh];
        a.y = arow[k + kh + 1];
        b.x = wcol[(long)(k + kh) * H_DIM];
        b.y = wcol[(long)(k + kh + 1) * H_DIM];
        acc = __builtin_amdgcn_wmma_f32_16x16x4_f32(
            /*neg_a=*/false, a, /*neg_b=*/false, b,
            /*c_mod=*/(short)0, acc, /*reuse_a=*/false, /*reuse_b=*/false);
    }

    // C/D layout: VGPR i -> row i (lanes 0-15) or row i+8 (lanes 16-31), col = lane&15
    const int  col   = nt * 16 + rc;
    const float bv   = bias[col];
    const long rbase = mt * 16 + ((lane >> 4) << 3);
    #pragma unroll
    for (int i = 0; i < 8; ++i) {
        float v = acc[i] + bv;
        out[(rbase + i) * H_DIM + col] = v > 0.0f ? v : 0.0f;
    }
}

extern "C" void kernel_launch(void* const* d_in, const int* in_sizes, int n_in,
                              void* d_out, int out_size, void* d_ws, size_t ws_size,
                              hipStream_t stream) {
    const float* x        = (const float*)d_in[0];
    const int*   inc_rows = (const int*)  d_in[1];
    const int*   inc_cols = (const int*)  d_in[2];
    const float* inc_vals = (const float*)d_in[3];
    const int*   y        = (const int*)  d_in[4];
    const int*   batch0   = (const int*)  d_in[5];
    const float* W0       = (const float*)d_in[6];
    const float* b0       = (const float*)d_in[7];
    const float* W1       = (const float*)d_in[8];
    const float* b1       = (const float*)d_in[9];
    float* out = (float*)d_out;

    const int N   = in_sizes[0] / C_DIM;   // 100000
    const int NNZ = in_sizes[1];           // 1600000
    const int E   = 50000;

    float* x1 = (float*)d_ws;              // [E, 128] accumulator, 25.6 MB scratch

    // 1) zero x1 + copy y/batch_0 as floats
    {
        long ec    = (long)E * C_DIM;
        long total = ec > (long)N ? ec : (long)N;
        int  thr   = 256;
        int  blk   = (int)((total + thr - 1) / thr);
        init_zero_copy<<<blk, thr, 0, stream>>>(x1, ec, y, batch0, out, N);
    }
    // 2) scatter-add gathered node features into hyperedges
    {
        int waves_per_block = 8;           // 256 threads = 8 wave32
        int blk = (NNZ + waves_per_block - 1) / waves_per_block;
        scatter_hyperedges<<<blk, 256, 0, stream>>>(x, inc_rows, inc_cols, inc_vals, x1, NNZ);
    }
    // 3) x0_out = relu(x @ W0 + b0)   -> d_out[2N .. 2N + N*H)
    {
        long tiles = (long)(N / 16) * (H_DIM / 16);
        int  blk   = (int)((tiles + 7) / 8);
        gemm_bias_relu<<<blk, 256, 0, stream>>>(x, W0, b0, out + 2L * N, N);
    }
    // 4) x1_out = relu(x1 @ W1 + b1)  -> d_out[2N + N*H .. end)
    {
        long tiles = (long)(E / 16) * (H_DIM / 16);
        int  blk   = (int)((tiles + 7) / 8);
        gemm_bias_relu<<<blk, 256, 0, stream>>>(x1, W1, b1,
                                                out + 2L * N + (long)N * H_DIM, E);
    }
}